// TGCN_4561255269273
// MI455X (gfx1250) — compile-verified
//
#include <hip/hip_runtime.h>
#include <hip/hip_bf16.h>

// ---------------------------------------------------------------------------
// TGCN on MI455X (gfx1250, wave32).
//  Pipeline: deg -> dinv -> self-loop init of agg -> edge scatter (f32 atomics,
//  F=16 features, all 16 timesteps per edge) -> WMMA f32 16x16x4 GEMM
//  [N,16]@[16,128] with per-column BN statistics -> pooled h_seq -> GRU -> MLP.
//  Note: mean-pool after train-mode BN equals bn_beta analytically; we compute
//  it from the WMMA-produced statistics as gamma*rsqrt(var+eps)*(mean-mu)+beta.
// ---------------------------------------------------------------------------

typedef __attribute__((ext_vector_type(2))) float v2f;
typedef __attribute__((ext_vector_type(8))) float v8f;

#define TSTEPS 16
#define FDIM   16
#define HDIM   128

// ---------------- degree / norm ----------------
__global__ void k_init_deg(float* deg, int N) {
    int i = blockIdx.x * blockDim.x + threadIdx.x;
    if (i < N) deg[i] = 1.0f;               // self loop
}

__global__ void k_deg_accum(const int* __restrict__ dst, float* deg, int E) {
    int e = blockIdx.x * blockDim.x + threadIdx.x;
    if (e < E) unsafeAtomicAdd(&deg[dst[e]], 1.0f);
}

__global__ void k_dinv(const float* __restrict__ deg, float* dinv, int N) {
    int i = blockIdx.x * blockDim.x + threadIdx.x;
    if (i < N) {
        float d = deg[i];
        dinv[i] = (d > 0.0f) ? rsqrtf(d) : 0.0f;
    }
}

// agg[t][i][f] = dinv[i]^2 * x[t][i][f]   (self-loop contribution == init)
__global__ void k_selfloop_init(const float* __restrict__ x,
                                const float* __restrict__ dinv,
                                float* agg, int N, int total) {
    int idx = blockIdx.x * blockDim.x + threadIdx.x;
    if (idx < total) {
        int node = (idx >> 4) % N;
        float w = dinv[node];
        agg[idx] = w * w * x[idx];
    }
}

__global__ void k_zero(float* p, int n) {
    int i = blockIdx.x * blockDim.x + threadIdx.x;
    if (i < n) p[i] = 0.0f;
}

// ---------------- edge scatter: agg[t][dst][f] += norm * x[t][src][f] -------
// One thread per (edge, feature); loop over all 16 timesteps so the edge
// metadata (src, dst, norm) is loaded once per edge.  x_seq (51MB), agg (51MB)
// and edge_index (12.8MB) are all L2-resident (192MB), so this phase is bound
// by L2 atomic/gather throughput, not HBM.
__global__ void k_edge_scatter(const int* __restrict__ ei,
                               const float* __restrict__ dinv,
                               const float* __restrict__ x,
                               float* agg, int E, int NF) {
    int gid = blockIdx.x * blockDim.x + threadIdx.x;
    if (gid >= E * FDIM) return;
    int e = gid >> 4;
    int f = gid & 15;
    int s = ei[e];
    int d = ei[E + e];
    float nm = dinv[s] * dinv[d];
    int xoff = s * FDIM + f;
    int aoff = d * FDIM + f;
#pragma unroll
    for (int t = 0; t < TSTEPS; ++t) {
        if (t + 1 < TSTEPS)
            __builtin_prefetch(&x[(t + 1) * NF + xoff], 0, 0);  // global_prefetch_b8
        float v = x[t * NF + xoff];
        unsafeAtomicAdd(&agg[t * NF + aoff], nm * v);
    }
}

// ---------------- WMMA GEMM + BN statistics ---------------------------------
// One wave32 per 16-row tile of agg[t].  y = agg_tile(16x16) @ W(16x128) + b,
// done as 8 column-tiles x 4 chained V_WMMA_F32_16X16X4_F32.
// A layout (32-bit 16x4): lane<16 -> M=lane, K={0,1}; lane>=16 -> K={2,3}.
// B layout mirrors A with N in the lane index.
// C/D: VGPR r holds row M=r (lanes 0-15) / M=r+8 (lanes 16-31), N = lane&15.
__global__ void k_gemm_stats(const float* __restrict__ agg,
                             const float* __restrict__ W,
                             const float* __restrict__ bias,
                             float* s1, float* s2,
                             int N, int tilesPerT) {
    int wid  = (blockIdx.x * blockDim.x + threadIdx.x) >> 5;  // global wave id
    int lane = threadIdx.x & 31;
    int totalTiles = TSTEPS * tilesPerT;
    if (wid >= totalTiles) return;                             // wave-uniform

    int t    = wid / tilesPerT;
    int tile = wid - t * tilesPerT;
    int row0 = tile * 16;
    bool fullTile = (row0 + 16 <= N);                          // wave-uniform

    const float* aggT = agg + (size_t)t * N * FDIM;
    int mrow = row0 + (lane & 15);
    int khi  = (lane >> 4) << 1;                               // 0 or 2
    bool rowOK = (mrow < N);

    v2f a[4];
#pragma unroll
    for (int q = 0; q < 4; ++q) {
        int k = q * 4 + khi;
        a[q].x = rowOK ? aggT[mrow * FDIM + k] : 0.0f;
        a[q].y = rowOK ? aggT[mrow * FDIM + k + 1] : 0.0f;
    }

    int colL  = lane & 15;
    int rbase = (lane >> 4) << 3;                              // 0 or 8

#pragma unroll
    for (int c = 0; c < 8; ++c) {
        int colg = c * 16 + colL;
        v2f b[4];
#pragma unroll
        for (int q = 0; q < 4; ++q) {
            int k = q * 4 + khi;
            b[q].x = W[k * HDIM + colg];
            b[q].y = W[(k + 1) * HDIM + colg];
        }
        v8f acc = {0.0f, 0.0f, 0.0f, 0.0f, 0.0f, 0.0f, 0.0f, 0.0f};
#pragma unroll
        for (int q = 0; q < 4; ++q) {
            // (neg_a, A, neg_b, B, c_mod, C, reuse_a, reuse_b)
            acc = __builtin_amdgcn_wmma_f32_16x16x4_f32(
                false, a[q], false, b[q], (short)0, acc, false, false);
        }
        float bb = bias[colg];
        float s = 0.0f, ss = 0.0f;
        if (fullTile) {
            // fast path: no per-row bounds predication (N % 16 == 0 case)
#pragma unroll
            for (int r = 0; r < 8; ++r) {
                float y = acc[r] + bb;
                s  += y;
                ss += y * y;
            }
        } else {
#pragma unroll
            for (int r = 0; r < 8; ++r) {
                if (row0 + rbase + r < N) {
                    float y = acc[r] + bb;
                    s  += y;
                    ss += y * y;
                }
            }
        }
        // lane L and L^16 cover the two row-halves of the same column
        s  += __shfl_xor(s, 16);
        ss += __shfl_xor(ss, 16);
        if (lane < 16) {
            unsafeAtomicAdd(&s1[t * HDIM + colg], s);
            unsafeAtomicAdd(&s2[t * HDIM + colg], ss);
        }
    }
}

// ---------------- pooled BN output ------------------------------------------
// mean over nodes of ((y-mu)*rsqrt(var+eps)*gamma + beta) with mu = mean(y).
__global__ void k_hseq(const float* __restrict__ s1, const float* __restrict__ s2,
                       const float* __restrict__ gamma, const float* __restrict__ beta,
                       float* hseq, int N) {
    int i = blockIdx.x * blockDim.x + threadIdx.x;
    if (i < TSTEPS * HDIM) {
        float invN = 1.0f / (float)N;
        float mean = s1[i] * invN;
        float var  = s2[i] * invN - mean * mean;
        float inv  = rsqrtf(var + 1e-5f);
        int h = i & (HDIM - 1);
        hseq[i] = (mean - mean) * inv * gamma[h] + beta[h];
    }
}

// ---------------- GRU (H=128, 16 steps) + output MLP, one block -------------
__global__ void k_gru_mlp(const float* __restrict__ hseq,
                          const float* __restrict__ w_ih, const float* __restrict__ w_hh,
                          const float* __restrict__ b_ih, const float* __restrict__ b_hh,
                          const float* __restrict__ w1, const float* __restrict__ b1,
                          const float* __restrict__ w2, const float* __restrict__ b2,
                          float* out) {
    __shared__ float hs[HDIM];
    __shared__ float xs[HDIM];
    __shared__ float h1s[HDIM / 2];
    int i = threadIdx.x;           // 0..127
    hs[i] = 0.0f;
    __syncthreads();

    for (int t = 0; t < TSTEPS; ++t) {
        xs[i] = hseq[t * HDIM + i];
        __syncthreads();
        float gxr = b_ih[i], gxz = b_ih[HDIM + i], gxn = b_ih[2 * HDIM + i];
        float ghr = b_hh[i], ghz = b_hh[HDIM + i], ghn = b_hh[2 * HDIM + i];
        const float* wr = w_ih + (size_t)i * HDIM;
        const float* wz = w_ih + (size_t)(HDIM + i) * HDIM;
        const float* wn = w_ih + (size_t)(2 * HDIM + i) * HDIM;
        const float* ur = w_hh + (size_t)i * HDIM;
        const float* uz = w_hh + (size_t)(HDIM + i) * HDIM;
        const float* un = w_hh + (size_t)(2 * HDIM + i) * HDIM;
#pragma unroll 4
        for (int k = 0; k < HDIM; ++k) {
            float xk = xs[k], hk = hs[k];
            gxr += wr[k] * xk;  gxz += wz[k] * xk;  gxn += wn[k] * xk;
            ghr += ur[k] * hk;  ghz += uz[k] * hk;  ghn += un[k] * hk;
        }
        float r = 1.0f / (1.0f + expf(-(gxr + ghr)));
        float z = 1.0f / (1.0f + expf(-(gxz + ghz)));
        float n = tanhf(gxn + r * ghn);
        float hn = (1.0f - z) * n + z * hs[i];
        __syncthreads();
        hs[i] = hn;
        __syncthreads();
    }

    // MLP: h1 = relu(h @ w1^T + b1); out = h1 @ w2^T + b2
    if (i < HDIM / 2) {
        float acc = b1[i];
        const float* row = w1 + (size_t)i * HDIM;
#pragma unroll 4
        for (int k = 0; k < HDIM; ++k) acc += row[k] * hs[k];
        h1s[i] = acc > 0.0f ? acc : 0.0f;
    }
    __syncthreads();
    if (i == 0) {
        float acc = b2[0];
        for (int j = 0; j < HDIM / 2; ++j) acc += w2[j] * h1s[j];
        out[0] = acc;
    }
}

// ---------------------------------------------------------------------------
extern "C" void kernel_launch(void* const* d_in, const int* in_sizes, int n_in,
                              void* d_out, int out_size, void* d_ws, size_t ws_size,
                              hipStream_t stream) {
    const float* x     = (const float*)d_in[0];
    const int*   ei    = (const int*)d_in[1];
    const float* gcn_w = (const float*)d_in[2];
    const float* gcn_b = (const float*)d_in[3];
    const float* gamma = (const float*)d_in[4];
    const float* beta  = (const float*)d_in[5];
    const float* w_ih  = (const float*)d_in[6];
    const float* w_hh  = (const float*)d_in[7];
    const float* b_ih  = (const float*)d_in[8];
    const float* b_hh  = (const float*)d_in[9];
    const float* w1    = (const float*)d_in[10];
    const float* b1    = (const float*)d_in[11];
    const float* w2    = (const float*)d_in[12];
    const float* b2    = (const float*)d_in[13];
    float* out = (float*)d_out;

    int E = in_sizes[1] / 2;
    int N = in_sizes[0] / (TSTEPS * FDIM);
    int NF = N * FDIM;

    // Workspace layout (floats): deg[N] | dinv[N] | agg[T*N*F] | s1[T*H] | s2[T*H] | hseq[T*H]
    float* ws   = (float*)d_ws;
    float* deg  = ws;
    float* dinv = ws + N;
    float* agg  = ws + 2 * (size_t)N;
    float* s1   = agg + (size_t)TSTEPS * NF;
    float* s2   = s1 + TSTEPS * HDIM;
    float* hseq = s2 + TSTEPS * HDIM;

    const int B = 256;
    k_init_deg<<<(N + B - 1) / B, B, 0, stream>>>(deg, N);
    k_deg_accum<<<(E + B - 1) / B, B, 0, stream>>>(ei + E, deg, E);
    k_dinv<<<(N + B - 1) / B, B, 0, stream>>>(deg, dinv, N);

    int totalAgg = TSTEPS * NF;
    k_selfloop_init<<<(totalAgg + B - 1) / B, B, 0, stream>>>(x, dinv, agg, N, totalAgg);
    k_zero<<<(2 * TSTEPS * HDIM + B - 1) / B, B, 0, stream>>>(s1, 2 * TSTEPS * HDIM);

    long long edgeThreads = (long long)E * FDIM;
    k_edge_scatter<<<(int)((edgeThreads + B - 1) / B), B, 0, stream>>>(
        ei, dinv, x, agg, E, NF);

    int tilesPerT = (N + 15) / 16;
    long long waves = (long long)TSTEPS * tilesPerT;
    k_gemm_stats<<<(int)((waves * 32 + B - 1) / B), B, 0, stream>>>(
        agg, gcn_w, gcn_b, s1, s2, N, tilesPerT);

    k_hseq<<<(TSTEPS * HDIM + B - 1) / B, B, 0, stream>>>(s1, s2, gamma, beta, hseq, N);

    k_gru_mlp<<<1, HDIM, 0, stream>>>(hseq, w_ih, w_hh, b_ih, b_hh,
                                      w1, b1, w2, b2, out);
    (void)n_in; (void)out_size; (void)ws_size;
}